// DynamicVoxelEncoder_446676599106
// MI455X (gfx1250) — compile-verified
//
#include <hip/hip_runtime.h>
#include <stdint.h>

// ---------------- problem constants (match reference) ----------------
#define BATCH     4
#define NFEAT     5
#define NX        1440
#define NY        1440
#define NZ        40
#define SENT      (NX * NY * NZ)          // 82,944,000
// hash table per batch
#define HASH_CAP  (1u << 19)              // 524288 slots, load factor <= 0.38
#define HASH_MASK (HASH_CAP - 1u)
#define EMPTY_KEY 0xFFFFFFFFu
// sort buffer per batch (power of two >= N=200000)
#define NSORT     (1u << 18)              // 262144
#define TILE_PTS  256                     // points per insert block
#define TILE_ELEM (TILE_PTS * NFEAT)      // 1280 floats = 5120 B
#define SORT_TILE 4096u                   // keys per LDS sort tile (16 KB)

// ---------------------------------------------------------------------
// init: clear hash tables, sort buffers, counters
__global__ __launch_bounds__(256) void k_init(uint32_t* __restrict__ keys,
                                              uint32_t* __restrict__ cnt,
                                              float*    __restrict__ sums,
                                              uint32_t* __restrict__ sortbuf,
                                              uint32_t* __restrict__ counters) {
    size_t tid = (size_t)blockIdx.x * blockDim.x + threadIdx.x;
    size_t tot = (size_t)BATCH * HASH_CAP;
    if (tid < tot) {
        keys[tid] = EMPTY_KEY;
        cnt[tid]  = 0u;
#pragma unroll
        for (int j = 0; j < NFEAT; ++j) sums[(size_t)j * tot + tid] = 0.0f;
    }
    if (tid < (size_t)BATCH * NSORT) sortbuf[tid] = EMPTY_KEY;
    if (tid < BATCH) counters[tid] = 0u;
}

// ---------------------------------------------------------------------
// insert: async-stage a 256-point tile (AoS, 5 floats/pt) into LDS with
// CDNA5 global_load_async_to_lds_b128 (512 B per wave-issue), wait on
// ASYNCcnt, then keep-test + voxel key + hash insert + atomic accumulate.
__global__ __launch_bounds__(256) void k_insert(const float* __restrict__ pts,
                                                const float* __restrict__ pc_range,
                                                const float* __restrict__ voxel,
                                                uint32_t* __restrict__ keys,
                                                uint32_t* __restrict__ cnt,
                                                float*    __restrict__ sums,
                                                int N) {
    __shared__ __align__(16) float tile[TILE_ELEM];
    const int b  = blockIdx.y;
    const int t  = threadIdx.x;
    const int p0 = blockIdx.x * TILE_PTS;

    const int rem   = N - p0;
    const int npts  = rem < TILE_PTS ? rem : TILE_PTS;
    const int total = npts * NFEAT;        // dwords to stage
    const int nvec  = total >> 2;          // 16-byte chunks (320 for full tile)
    const int tail  = total & 3;           // leftover dwords

    const float* gbase = pts + ((size_t)b * N + p0) * NFEAT;   // 16B-aligned
    const uint32_t lds0 = (uint32_t)(uintptr_t)(&tile[0]);

    // b128 async copies: 2 iterations cover 512 >= 320 chunks
#pragma unroll
    for (int i = 0; i < 2; ++i) {
        int e4 = i * TILE_PTS + t;
        if (e4 < nvec) {
            uint64_t gaddr   = (uint64_t)(uintptr_t)(gbase + (size_t)e4 * 4);
            uint32_t ldsaddr = lds0 + (uint32_t)e4 * 16u;
            asm volatile("global_load_async_to_lds_b128 %0, %1, off"
                         :: "v"(ldsaddr), "v"(gaddr)
                         : "memory");
        }
    }
    if (t < tail) {   // non-16B remainder (only for npts % 4 != 0)
        int e = (nvec << 2) + t;
        uint64_t gaddr   = (uint64_t)(uintptr_t)(gbase + e);
        uint32_t ldsaddr = lds0 + (uint32_t)e * 4u;
        asm volatile("global_load_async_to_lds_b32 %0, %1, off"
                     :: "v"(ldsaddr), "v"(gaddr)
                     : "memory");
    }
    asm volatile("s_wait_asynccnt 0" ::: "memory");
    __syncthreads();

    if (p0 + t >= N) return;

    const float r0 = pc_range[0], r1 = pc_range[1], r2 = pc_range[2];
    const float r3 = pc_range[3], r4 = pc_range[4], r5 = pc_range[5];
    const float vx = voxel[0], vy = voxel[1], vz = voxel[2];

    const float x  = tile[t * NFEAT + 0];
    const float y  = tile[t * NFEAT + 1];
    const float z  = tile[t * NFEAT + 2];
    const float f3 = tile[t * NFEAT + 3];
    const float f4 = tile[t * NFEAT + 4];

    const bool keep = (x >= r0) && (x <= r3) && (y >= r1) && (y <= r4) &&
                      (z >= r2) && (z <= r5);
    if (!keep) return;

    const int cx = (int)((x - r0) / vx);   // trunc toward zero == astype(int32)
    const int cy = (int)((y - r1) / vy);
    const int cz = (int)((z - r2) / vz);
    const uint32_t key = (uint32_t)(cz * (NY * NX) + cy * NX + cx);

    uint32_t* K = keys + (size_t)b * HASH_CAP;
    uint32_t slot = (key * 2654435761u) & HASH_MASK;
    for (;;) {
        uint32_t prev = atomicCAS(&K[slot], EMPTY_KEY, key);
        if (prev == EMPTY_KEY || prev == key) break;
        slot = (slot + 1u) & HASH_MASK;
    }
    const size_t tot = (size_t)BATCH * HASH_CAP;
    const size_t idx = (size_t)b * HASH_CAP + slot;
    atomicAdd(&cnt[idx], 1u);
    float vals[NFEAT] = { x, y, z, f3, f4 };
#pragma unroll
    for (int j = 0; j < NFEAT; ++j)
        atomicAdd(&sums[(size_t)j * tot + idx], vals[j]);
}

// ---------------------------------------------------------------------
// compact occupied hash slots into the sort buffer
__global__ __launch_bounds__(256) void k_compact(const uint32_t* __restrict__ keys,
                                                 uint32_t* __restrict__ sortbuf,
                                                 uint32_t* __restrict__ counters) {
    const int b = blockIdx.y;
    const uint32_t s = blockIdx.x * 256u + threadIdx.x;
    if (s >= HASH_CAP) return;
    uint32_t key = keys[(size_t)b * HASH_CAP + s];
    if (key != EMPTY_KEY) {
        uint32_t idx = atomicAdd(&counters[b], 1u);
        sortbuf[(size_t)b * NSORT + idx] = key;
    }
}

// ---------------------------------------------------------------------
// bitonic sort: 4096-element LDS tiles (16 KB), 256 threads, 8 pairs/step
__global__ __launch_bounds__(256) void k_sort_local_init(uint32_t* __restrict__ sortbuf) {
    __shared__ uint32_t s[SORT_TILE];
    uint32_t* buf = sortbuf + (size_t)blockIdx.y * NSORT;
    const unsigned base = blockIdx.x * SORT_TILE;
    const unsigned t = threadIdx.x;
    for (unsigned i = t; i < SORT_TILE; i += 256u) s[i] = buf[base + i];
    __syncthreads();
    for (unsigned k = 2u; k <= SORT_TILE; k <<= 1) {
        for (unsigned j = k >> 1; j > 0u; j >>= 1) {
            for (unsigned p = t; p < SORT_TILE / 2u; p += 256u) {
                unsigned i = ((p & ~(j - 1u)) << 1) | (p & (j - 1u));
                bool asc = (((base + i) & k) == 0u);
                uint32_t a = s[i], c = s[i + j];
                if ((a > c) == asc) { s[i] = c; s[i + j] = a; }
            }
            __syncthreads();
        }
    }
    for (unsigned i = t; i < SORT_TILE; i += 256u) buf[base + i] = s[i];
}

__global__ __launch_bounds__(256) void k_sort_global(uint32_t* __restrict__ sortbuf,
                                                     unsigned k, unsigned j) {
    uint32_t* buf = sortbuf + (size_t)blockIdx.y * NSORT;
    unsigned p = blockIdx.x * 256u + threadIdx.x;          // pair id < NSORT/2
    unsigned i = ((p & ~(j - 1u)) << 1) | (p & (j - 1u));
    bool asc = ((i & k) == 0u);
    uint32_t a = buf[i], c = buf[i + j];
    if ((a > c) == asc) { buf[i] = c; buf[i + j] = a; }
}

__global__ __launch_bounds__(256) void k_sort_local_final(uint32_t* __restrict__ sortbuf,
                                                          unsigned k) {
    __shared__ uint32_t s[SORT_TILE];
    uint32_t* buf = sortbuf + (size_t)blockIdx.y * NSORT;
    const unsigned base = blockIdx.x * SORT_TILE;
    const unsigned t = threadIdx.x;
    for (unsigned i = t; i < SORT_TILE; i += 256u) s[i] = buf[base + i];
    __syncthreads();
    for (unsigned j = SORT_TILE / 2u; j > 0u; j >>= 1) {
        for (unsigned p = t; p < SORT_TILE / 2u; p += 256u) {
            unsigned i = ((p & ~(j - 1u)) << 1) | (p & (j - 1u));
            bool asc = (((base + i) & k) == 0u);
            uint32_t a = s[i], c = s[i + j];
            if ((a > c) == asc) { s[i] = c; s[i + j] = a; }
        }
        __syncthreads();
    }
    for (unsigned i = t; i < SORT_TILE; i += 256u) buf[base + i] = s[i];
}

// ---------------------------------------------------------------------
// output: sorted unique keys -> mean features + (b,z,y,x); pad rows 0/-1
__global__ __launch_bounds__(256) void k_output(const uint32_t* __restrict__ sortbuf,
                                                const uint32_t* __restrict__ keys,
                                                const uint32_t* __restrict__ cnt,
                                                const float*    __restrict__ sums,
                                                float* __restrict__ vox,
                                                int*   __restrict__ coors,
                                                int*   __restrict__ grid,
                                                int N) {
    const int b = blockIdx.y;
    const int i = blockIdx.x * 256 + threadIdx.x;
    if (i >= N) return;
    if (b == 0 && i == 0) { grid[0] = NX; grid[1] = NY; grid[2] = NZ; }

    const uint32_t key = sortbuf[(size_t)b * NSORT + i];
    const size_t row = (size_t)b * N + i;
    if (key != EMPTY_KEY) {
        const uint32_t* K = keys + (size_t)b * HASH_CAP;
        uint32_t slot = (key * 2654435761u) & HASH_MASK;
        while (K[slot] != key) slot = (slot + 1u) & HASH_MASK;
        const size_t tot = (size_t)BATCH * HASH_CAP;
        const size_t idx = (size_t)b * HASH_CAP + slot;
        const float c = (float)cnt[idx];
        const float inv = 1.0f / fmaxf(c, 1.0f);
#pragma unroll
        for (int j = 0; j < NFEAT; ++j)
            vox[row * NFEAT + j] = sums[(size_t)j * tot + idx] * inv;
        const bool valid = key < (uint32_t)SENT;
        const int zc = (int)(key / (uint32_t)(NX * NY));
        const int rm = (int)(key % (uint32_t)(NX * NY));
        coors[row * 4 + 0] = b;
        coors[row * 4 + 1] = valid ? zc : -1;
        coors[row * 4 + 2] = valid ? (rm / NX) : -1;
        coors[row * 4 + 3] = valid ? (rm % NX) : -1;
    } else {
#pragma unroll
        for (int j = 0; j < NFEAT; ++j) vox[row * NFEAT + j] = 0.0f;
        coors[row * 4 + 0] = b;
        coors[row * 4 + 1] = -1;
        coors[row * 4 + 2] = -1;
        coors[row * 4 + 3] = -1;
    }
}

// ---------------------------------------------------------------------
extern "C" void kernel_launch(void* const* d_in, const int* in_sizes, int n_in,
                              void* d_out, int out_size, void* d_ws, size_t ws_size,
                              hipStream_t stream) {
    const float* pts = (const float*)d_in[0];
    const float* pr  = (const float*)d_in[1];
    const float* vs  = (const float*)d_in[2];
    const int N = in_sizes[0] / (BATCH * NFEAT);   // 200000

    // workspace layout
    const size_t tot = (size_t)BATCH * HASH_CAP;
    uint32_t* keys     = (uint32_t*)d_ws;                         // tot u32
    uint32_t* cnt      = keys + tot;                              // tot u32
    float*    sums     = (float*)(cnt + tot);                     // 5*tot f32
    uint32_t* sortbuf  = (uint32_t*)(sums + (size_t)NFEAT * tot); // B*NSORT u32
    uint32_t* counters = sortbuf + (size_t)BATCH * NSORT;         // B u32

    float* vox  = (float*)d_out;
    int*   coor = (int*)d_out + (size_t)BATCH * N * NFEAT;
    int*   grid = coor + (size_t)BATCH * N * 4;

    k_init<<<dim3((unsigned)((tot + 255) / 256)), 256, 0, stream>>>(
        keys, cnt, sums, sortbuf, counters);
    k_insert<<<dim3((N + TILE_PTS - 1) / TILE_PTS, BATCH), TILE_PTS, 0, stream>>>(
        pts, pr, vs, keys, cnt, sums, N);
    k_compact<<<dim3(HASH_CAP / 256u, BATCH), 256, 0, stream>>>(keys, sortbuf, counters);

    // bitonic sort: strides <= 2048 fused in 4096-key LDS tiles,
    // global passes only for strides >= 4096 (21 passes total)
    k_sort_local_init<<<dim3(NSORT / SORT_TILE, BATCH), 256, 0, stream>>>(sortbuf);
    for (unsigned k = 2u * SORT_TILE; k <= NSORT; k <<= 1) {
        for (unsigned j = k >> 1; j >= SORT_TILE; j >>= 1)
            k_sort_global<<<dim3(NSORT / 2u / 256u, BATCH), 256, 0, stream>>>(sortbuf, k, j);
        k_sort_local_final<<<dim3(NSORT / SORT_TILE, BATCH), 256, 0, stream>>>(sortbuf, k);
    }

    k_output<<<dim3((N + 255) / 256, BATCH), 256, 0, stream>>>(
        sortbuf, keys, cnt, sums, vox, coor, grid, N);
}